// GraphSpectralFilterLayer_8796093022366
// MI455X (gfx1250) — compile-verified
//
#include <hip/hip_runtime.h>
#include <hip/hip_bf16.h>
#include <math.h>

// ---------------- problem constants (match reference) ----------------
#define IN_F   512
#define OUT_F  128
#define CCH    4
#define ALPHA  0.2f
#define BIGF   9e15f

typedef __attribute__((ext_vector_type(2))) float v2f;
typedef __attribute__((ext_vector_type(8))) float v8f;

// ---------------- kernel 0: zero fill (float4 wide stores) ----------------
__global__ void zero_f4_kernel(float4* __restrict__ p, size_t n4) {
    size_t i = (size_t)blockIdx.x * blockDim.x + threadIdx.x;
    size_t stride = (size_t)gridDim.x * blockDim.x;
    const float4 z = make_float4(0.f, 0.f, 0.f, 0.f);
    for (; i < n4; i += stride) p[i] = z;
}

// ---------------- kernel 1: edge transform + divisor segment-sum ----------------
// v = min(exp(mask(leaky_relu(ev))), BIG);  divisor[c][row] += v
__global__ void edge_transform_kernel(const float* __restrict__ ev,
                                      const int* __restrict__ rowi,
                                      float* __restrict__ vout,
                                      float* __restrict__ divisor,
                                      int E, int N) {
    int i = blockIdx.x * blockDim.x + threadIdx.x;   // over C*E
    if (i >= CCH * E) return;
    int c = i / E;
    int e = i - c * E;
    float u  = ev[i];
    float lr = (u > 0.0f) ? u : ALPHA * u;           // leaky_relu (NaN stays NaN)
    if (isnan(lr) || lr == 0.0f) lr = -BIGF;
    float vv = fminf(expf(lr), BIGF);                // exp(-BIG) underflows to 0
    vout[i] = vv;
    atomicAdd(divisor + c * N + rowi[e], vv);
}

// ---------------- kernel 2: h = x @ W via V_WMMA_F32_16X16X4_F32 ----------------
// One wave32 computes one 16x16 fp32 output tile; K-loop in steps of 4.
// A (16x4 f32): lanes 0-15 hold M=0..15; VGPR0/1 = K+koff, K+koff+1 with
//   koff = 2*(lane>>4).  B (4x16) mirrored with M<->N.
__global__ void gemm_h_wmma_kernel(const float* __restrict__ x,
                                   const float* __restrict__ W,
                                   float* __restrict__ h, int N) {
    int gtid = blockIdx.x * blockDim.x + threadIdx.x;
    int wave = gtid >> 5;
    int lane = threadIdx.x & 31;
    const int tilesN = OUT_F / 16;                   // 8
    int tm = wave / tilesN;
    int tn = wave - tm * tilesN;
    int row0 = tm * 16;
    int col0 = tn * 16;
    if (row0 >= N) return;                           // wave-uniform (EXEC stays all-1s)

    int mn   = lane & 15;                            // M index for A, N index for B
    int koff = (lane >> 4) << 1;                     // 0 or 2

    const float* xrow = x + (size_t)(row0 + mn) * IN_F;
    v8f acc = {};
    for (int k = 0; k < IN_F; k += 4) {
        v2f a, b;
        a.x = xrow[k + koff];
        a.y = xrow[k + koff + 1];
        b.x = W[(size_t)(k + koff)     * OUT_F + col0 + mn];
        b.y = W[(size_t)(k + koff + 1) * OUT_F + col0 + mn];
        // (neg_a, A, neg_b, B, c_mod, C, reuse_a, reuse_b)
        acc = __builtin_amdgcn_wmma_f32_16x16x4_f32(false, a, false, b,
                                                    (short)0, acc, false, false);
    }
    // C/D 16x16 f32 layout: VGPR r, lanes 0-15 -> M=r, lanes 16-31 -> M=r+8; N=lane&15
    int mbase = row0 + ((lane >> 4) << 3);
    int ncol  = col0 + mn;
#pragma unroll
    for (int r = 0; r < 8; ++r)
        h[(size_t)(mbase + r) * OUT_F + ncol] = acc[r];
}

// ---------------- kernel 3: edge scatter (normalized) ----------------
// s = v / max(divisor,1-if-zero); agg[c][row][f] += s * h[col][f];
// A[c][row][col] += s  (dense attention output region)
#define EDGES_PER_BLOCK 8
__global__ void edge_scatter_kernel(const int* __restrict__ rowi,
                                    const int* __restrict__ coli,
                                    const float* __restrict__ v,
                                    const float* __restrict__ divisor,
                                    const float* __restrict__ h,
                                    float* __restrict__ agg,
                                    float* __restrict__ Aout,
                                    int E, int N) {
    int tid = threadIdx.x;                           // 0..127 feature lane
    int e0 = blockIdx.x * EDGES_PER_BLOCK;
    for (int j = 0; j < EDGES_PER_BLOCK; ++j) {
        int e = e0 + j;
        if (e >= E) break;
        int r  = rowi[e];
        int cl = coli[e];
        float hv = h[(size_t)cl * OUT_F + tid];      // reused for all 4 channels
#pragma unroll
        for (int c = 0; c < CCH; ++c) {
            float vv = v[(size_t)c * E + e];
            float d  = divisor[c * N + r];
            float s  = vv / ((d == 0.0f) ? 1.0f : d);
            atomicAdd(&agg[((size_t)c * N + r) * OUT_F + tid], s * hv);
            if (tid == 0)
                atomicAdd(&Aout[(size_t)c * N * N + (size_t)r * N + cl], s);
        }
    }
}

// ---------------- kernel 4: elu finalize into [n, c*128+f] layout ----------------
__global__ void finalize_kernel(const float* __restrict__ agg,
                                float* __restrict__ out, int N) {
    int i = blockIdx.x * blockDim.x + threadIdx.x;   // over N*C*OUT_F
    int total = N * CCH * OUT_F;
    if (i >= total) return;
    int f = i & (OUT_F - 1);
    int t = i >> 7;                                  // = n*CCH + c
    int c = t & (CCH - 1);
    int n = t >> 2;
    float a = agg[((size_t)c * N + n) * OUT_F + f];
    out[i] = (a > 0.0f) ? a : (expf(a) - 1.0f);      // elu(alpha=1)
}

// ---------------- launcher ----------------
extern "C" void kernel_launch(void* const* d_in, const int* in_sizes, int n_in,
                              void* d_out, int out_size, void* d_ws, size_t ws_size,
                              hipStream_t stream) {
    const float* x  = (const float*)d_in[0];   // [N, IN_F]
    const float* W  = (const float*)d_in[1];   // [IN_F, OUT_F]
    const float* ev = (const float*)d_in[2];   // [C, E]
    const int* eidx = (const int*)d_in[3];     // [2, E]

    const int N = in_sizes[0] / IN_F;          // 4096
    const int E = in_sizes[3] / 2;             // 262144
    const int* rowi = eidx;
    const int* coli = eidx + E;

    // workspace layout (floats)
    float* ws      = (float*)d_ws;
    float* ws_v    = ws;                               // C*E
    float* ws_h    = ws_v + (size_t)CCH * E;           // N*OUT_F
    float* ws_div  = ws_h + (size_t)N * OUT_F;         // C*N
    float* ws_agg  = ws_div + (size_t)CCH * N;         // C*N*OUT_F

    // output layout: [N, C*OUT_F] h' block, then [C, N, N] attention block
    float* out_h = (float*)d_out;
    float* out_A = out_h + (size_t)N * CCH * OUT_F;

    // 0a) zero dense attention region (C*N*N floats; 268 MB — the HBM floor)
    size_t nA4 = ((size_t)CCH * N * N) / 4;
    zero_f4_kernel<<<8192, 256, 0, stream>>>((float4*)out_A, nA4);
    // 0b) zero divisor + agg (contiguous in ws)
    size_t nW4 = ((size_t)CCH * N + (size_t)CCH * N * OUT_F) / 4;
    zero_f4_kernel<<<2048, 256, 0, stream>>>((float4*)ws_div, nW4);

    // 1) edge transform + divisor segment-sum
    {
        int total = CCH * E;
        edge_transform_kernel<<<(total + 255) / 256, 256, 0, stream>>>(
            ev, rowi, ws_v, ws_div, E, N);
    }

    // 2) h = x @ W  (fp32 WMMA, exact precision; <3% of runtime)
    {
        int tiles = (N / 16) * (OUT_F / 16);   // waves
        int blocks = (tiles + 7) / 8;          // 8 waves / 256-thread block
        gemm_h_wmma_kernel<<<blocks, 256, 0, stream>>>(x, W, ws_h, N);
    }

    // 3) normalized scatter: dense A + agg (agg/h stay L2-resident)
    {
        int blocks = (E + EDGES_PER_BLOCK - 1) / EDGES_PER_BLOCK;
        edge_scatter_kernel<<<blocks, OUT_F, 0, stream>>>(
            rowi, coli, ws_v, ws_div, ws_h, ws_agg, out_A, E, N);
    }

    // 4) elu + layout transform into output
    {
        int total = N * CCH * OUT_F;
        finalize_kernel<<<(total + 255) / 256, 256, 0, stream>>>(ws_agg, out_h, N);
    }
}